// MPNNNet_12867722019659
// MI455X (gfx1250) — compile-verified
//
#include <hip/hip_runtime.h>
#include <math.h>

typedef float v2f __attribute__((ext_vector_type(2)));
typedef float v8f __attribute__((ext_vector_type(8)));

// ---------------------------------------------------------------------------
// Pack W[K,N] -> Wp so each WMMA B-fragment is one aligned b64 load:
//   Wp[(r*N + c)*2 + j] = W[(2r + j)*N + c],  r in [0,K/2), j in {0,1}
// For k-step (k%4==0), lane-half hi: fragment base = ((k/2 + hi)*N + col)*2.
// Advancing k by 4 advances the index by 4N floats (pointer bump).
// ---------------------------------------------------------------------------
__global__ void pack_w_pairs(const float* __restrict__ W, float* __restrict__ Wp,
                             int N, long long total) {
  long long i = (long long)blockIdx.x * blockDim.x + threadIdx.x;
  if (i >= total) return;
  const int j = (int)(i & 1);
  const long long rc = i >> 1;
  const int c = (int)(rc % N);
  const long long r = rc / N;
  Wp[i] = W[(size_t)(2 * r + j) * N + c];
}

// ---------------------------------------------------------------------------
// GEMM with bias: C[M,N] = A[M,K] @ W[K,N] + bias[N]
// One wave computes a (16*MT) x 64 strip with V_WMMA_F32_16X16X4_F32,
// software-pipelined: next K-step fragments are loaded before current WMMAs.
// ---------------------------------------------------------------------------
template <int MT>   // 16-row tiles per wave (1 or 2)
__global__ void gemm_bias_wmma(const float* __restrict__ A,
                               const float* __restrict__ Wp,  // packed weights
                               const float* __restrict__ bias,
                               float* __restrict__ C,
                               int mStart, int M, int N, int K) {
  const int lane = threadIdx.x & 31;
  const int wave = (blockIdx.x * blockDim.x + threadIdx.x) >> 5;
  const int nGroups = N >> 6;                       // groups of 64 columns
  const int strip  = wave / nGroups;
  const int nGroup = wave - strip * nGroups;
  const int m0 = mStart + strip * (16 * MT);
  if (m0 >= M) return;                              // wave-uniform

  const int l15 = lane & 15;
  const int hi  = lane >> 4;                        // 0 or 1
  const int col0 = (nGroup << 6) + l15;

  v8f acc[4 * MT];
#pragma unroll
  for (int t = 0; t < 4; ++t) {
    const float bv = bias[col0 + t * 16];
#pragma unroll
    for (int m = 0; m < MT; ++m)
#pragma unroll
      for (int r = 0; r < 8; ++r) acc[m * 4 + t][r] = bv;
  }

  // A: lane holds row m0+l15 (+16 for second tile), columns k+2*hi, k+2*hi+1
  const float* ak0 = A + (size_t)(m0 + l15) * K + 2 * hi;
  const float* ak1 = ak0 + (size_t)16 * K;          // used when MT==2
  // B: packed fragment stream
  const float* wk = Wp + ((size_t)hi * N + col0) * 2;

  // ---- prologue: load fragments for k=0
  v2f a0 = *(const v2f*)ak0;
  v2f a1 = {};
  if (MT == 2) a1 = *(const v2f*)ak1;
  v2f b[4];
#pragma unroll
  for (int t = 0; t < 4; ++t) b[t] = *(const v2f*)(wk + t * 32);

  const int steps = K >> 2;
  for (int s = 0; s < steps - 1; ++s) {
    // ---- issue next-step loads first (hide latency behind 4*MT WMMAs)
    const float* wn = wk + (size_t)4 * N;
    v2f a0n = *(const v2f*)(ak0 + 4);
    v2f a1n = {};
    if (MT == 2) a1n = *(const v2f*)(ak1 + 4);
    v2f bn[4];
#pragma unroll
    for (int t = 0; t < 4; ++t) bn[t] = *(const v2f*)(wn + t * 32);

    // ---- current-step WMMAs
#pragma unroll
    for (int t = 0; t < 4; ++t) {
      acc[t] = __builtin_amdgcn_wmma_f32_16x16x4_f32(
          false, a0, false, b[t], (short)0, acc[t], false, false);
      if (MT == 2)
        acc[4 + t] = __builtin_amdgcn_wmma_f32_16x16x4_f32(
            false, a1, false, b[t], (short)0, acc[4 + t], false, false);
    }
    a0 = a0n; a1 = a1n;
#pragma unroll
    for (int t = 0; t < 4; ++t) b[t] = bn[t];
    wk = wn; ak0 += 4; ak1 += 4;
  }
  // ---- epilogue WMMAs (no more loads)
#pragma unroll
  for (int t = 0; t < 4; ++t) {
    acc[t] = __builtin_amdgcn_wmma_f32_16x16x4_f32(
        false, a0, false, b[t], (short)0, acc[t], false, false);
    if (MT == 2)
      acc[4 + t] = __builtin_amdgcn_wmma_f32_16x16x4_f32(
          false, a1, false, b[t], (short)0, acc[4 + t], false, false);
  }

  // ---- store: C/D layout N = lane&15, M = r + 8*hi (+16 per extra tile)
#pragma unroll
  for (int m = 0; m < MT; ++m) {
    const int mBase = m0 + m * 16 + hi * 8;
#pragma unroll
    for (int t = 0; t < 4; ++t) {
      float* cCol = C + (size_t)mBase * N + col0 + t * 16;
#pragma unroll
      for (int r = 0; r < 8; ++r) cCol[(size_t)r * N] = acc[m * 4 + t][r];
    }
  }
}

// ---------------------------------------------------------------------------
// agg init (self-loop message): vectorized copy
// ---------------------------------------------------------------------------
__global__ void copy_f4(const float4* __restrict__ s, float4* __restrict__ d,
                        long long n4) {
  long long i = (long long)blockIdx.x * blockDim.x + threadIdx.x;
  if (i < n4) d[i] = s[i];
}

__global__ void fill_ones(float* __restrict__ p, int n) {
  int i = blockIdx.x * blockDim.x + threadIdx.x;
  if (i < n) p[i] = 1.0f;
}

// In-degree counts (self loop already accounted by fill_ones)
__global__ void count_edges(const int* __restrict__ dIdx,
                            float* __restrict__ cnt, int E) {
  int i = blockIdx.x * blockDim.x + threadIdx.x;
  if (i < E) {
    __hip_atomic_fetch_add(&cnt[dIdx[i]], 1.0f,
                           __ATOMIC_RELAXED, __HIP_MEMORY_SCOPE_AGENT);
  }
}

// ---------------------------------------------------------------------------
// Edge scatter: agg[dst] += h[src]; one thread per float4 feature chunk.
// Consecutive threads walk the same source row -> coalesced float4 loads and
// coalesced global_atomic_add_f32 to the destination row (resolved in L2).
// ---------------------------------------------------------------------------
__global__ void scatter_edges(const int* __restrict__ sIdx,
                              const int* __restrict__ dIdx,
                              const float* __restrict__ h,
                              float* __restrict__ agg,
                              int F, int chunkShift, long long nWork) {
  long long gid = (long long)blockIdx.x * blockDim.x + threadIdx.x;
  if (gid >= nWork) return;
  const int e = (int)(gid >> chunkShift);
  const int c = (int)(gid & ((1 << chunkShift) - 1));
  const int s = sIdx[e];
  const int d = dIdx[e];
  const float4 v = *(const float4*)(h + (size_t)s * F + (c << 2));
  float* out = agg + (size_t)d * F + (c << 2);
  __hip_atomic_fetch_add(out + 0, v.x, __ATOMIC_RELAXED, __HIP_MEMORY_SCOPE_AGENT);
  __hip_atomic_fetch_add(out + 1, v.y, __ATOMIC_RELAXED, __HIP_MEMORY_SCOPE_AGENT);
  __hip_atomic_fetch_add(out + 2, v.z, __ATOMIC_RELAXED, __HIP_MEMORY_SCOPE_AGENT);
  __hip_atomic_fetch_add(out + 3, v.w, __ATOMIC_RELAXED, __HIP_MEMORY_SCOPE_AGENT);
}

// mean + ReLU fused: out[i] = max(agg[i]/cnt[row], 0)
__global__ void mean_relu(const float* __restrict__ agg,
                          const float* __restrict__ cnt,
                          float* __restrict__ out, int shiftF, long long n) {
  long long i = (long long)blockIdx.x * blockDim.x + threadIdx.x;
  if (i >= n) return;
  const int row = (int)(i >> shiftF);
  out[i] = fmaxf(agg[i] / cnt[row], 0.0f);
}

// ---------------------------------------------------------------------------
// mean + log_softmax over 64 classes: one wave32 per node, 2 values per lane.
// ---------------------------------------------------------------------------
__global__ void mean_logsoftmax64(const float* __restrict__ agg,
                                  const float* __restrict__ cnt,
                                  float* __restrict__ out, int M) {
  const int wave = (blockIdx.x * blockDim.x + threadIdx.x) >> 5;
  const int lane = threadIdx.x & 31;
  if (wave >= M) return;
  const float invc = 1.0f / cnt[wave];
  const size_t base = (size_t)wave * 64;
  const float v0 = agg[base + lane] * invc;
  const float v1 = agg[base + 32 + lane] * invc;

  float m = fmaxf(v0, v1);
#pragma unroll
  for (int off = 16; off; off >>= 1) m = fmaxf(m, __shfl_xor(m, off, 32));
  float s = expf(v0 - m) + expf(v1 - m);
#pragma unroll
  for (int off = 16; off; off >>= 1) s += __shfl_xor(s, off, 32);
  const float lse = m + logf(s);

  out[base + lane]      = v0 - lse;
  out[base + 32 + lane] = v1 - lse;
}

// ---------------------------------------------------------------------------
// Launch: packW -> GEMM1 -> self-loop init -> count -> scatter(256)
//         -> mean+ReLU -> GEMM2 -> self-loop init -> scatter(64)
//         -> mean+log_softmax
// ---------------------------------------------------------------------------
extern "C" void kernel_launch(void* const* d_in, const int* in_sizes, int n_in,
                              void* d_out, int out_size, void* d_ws, size_t ws_size,
                              hipStream_t stream) {
  const float* x  = (const float*)d_in[0];
  const int*   ei = (const int*)d_in[1];
  const float* W1 = (const float*)d_in[2];
  const float* b1 = (const float*)d_in[3];
  const float* W2 = (const float*)d_in[4];
  const float* b2 = (const float*)d_in[5];
  float* out = (float*)d_out;

  const int NF = 512, NH = 256, NC = 64;
  const int M = in_sizes[0] / NF;              // 50000 nodes
  const int E = in_sizes[1] / 2;               // 1.6M edges
  const int* srcI = ei;
  const int* dstI = ei + E;

  // workspace layout (bytes, all regions 256B aligned)
  char* ws = (char*)d_ws;
  const size_t sz_h1  = (size_t)M * NH * sizeof(float);               // 51.2 MB
  const size_t sz_h2  = (size_t)M * NC * sizeof(float);               // 12.8 MB
  const size_t sz_cnt = (((size_t)M * sizeof(float)) + 255) & ~(size_t)255;
  float* h1   = (float*)(ws);                                          // + relu(mean)
  float* agg1 = (float*)(ws + sz_h1);
  float* cnt  = (float*)(ws + 2 * sz_h1);
  float* h2   = (float*)(ws + 2 * sz_h1 + sz_cnt);
  float* agg2 = (float*)(ws + 2 * sz_h1 + sz_cnt + sz_h2);
  float* Wp1  = (float*)(ws + 2 * sz_h1 + sz_cnt + 2 * sz_h2);         // 512 KB
  float* Wp2  = Wp1 + (size_t)NF * NH;                                 // 64 KB

  // ---- pack weights into fragment-paired layout (tiny)
  {
    const long long t1 = (long long)NF * NH;
    pack_w_pairs<<<(unsigned)((t1 + 255) / 256), 256, 0, stream>>>(W1, Wp1, NH, t1);
    const long long t2 = (long long)NH * NC;
    pack_w_pairs<<<(unsigned)((t2 + 255) / 256), 256, 0, stream>>>(W2, Wp2, NC, t2);
  }

  // ---- Layer 1: h1 = x @ W1 + b1 (WMMA f32, 32-row strips + 16-row tail)
  {
    const int strips2 = M / 32;                    // full 32-row strips
    const int remRows = M - strips2 * 32;          // 0 or 16
    const int ng = NH / 64;
    if (strips2 > 0) {
      const int waves = strips2 * ng;
      gemm_bias_wmma<2><<<(waves + 3) / 4, 128, 0, stream>>>(
          x, Wp1, b1, h1, 0, strips2 * 32, NH, NF);
    }
    if (remRows > 0) {
      gemm_bias_wmma<1><<<1, 32 * ng, 0, stream>>>(
          x, Wp1, b1, h1, strips2 * 32, M, NH, NF);
    }
  }
  // ---- agg1 = h1 (self loop), cnt = 1 + indegree
  {
    const long long n4 = (long long)M * NH / 4;
    copy_f4<<<(unsigned)((n4 + 255) / 256), 256, 0, stream>>>(
        (const float4*)h1, (float4*)agg1, n4);
    fill_ones<<<(M + 255) / 256, 256, 0, stream>>>(cnt, M);
    count_edges<<<(E + 255) / 256, 256, 0, stream>>>(dstI, cnt, E);
  }
  // ---- agg1 += h1[src] over edges (256 feats -> 64 float4 chunks/edge)
  {
    const long long nW = (long long)E * (NH / 4);
    scatter_edges<<<(unsigned)((nW + 255) / 256), 256, 0, stream>>>(
        srcI, dstI, h1, agg1, NH, 6, nW);
  }
  // ---- h1 = relu(agg1 / cnt)
  {
    const long long n = (long long)M * NH;
    mean_relu<<<(unsigned)((n + 255) / 256), 256, 0, stream>>>(agg1, cnt, h1, 8, n);
  }
  // ---- Layer 2: h2 = h1 @ W2 + b2 (WMMA f32)
  {
    const int strips2 = M / 32;
    const int remRows = M - strips2 * 32;
    const int ng = NC / 64;                        // == 1
    if (strips2 > 0) {
      const int waves = strips2 * ng;
      gemm_bias_wmma<2><<<(waves + 3) / 4, 128, 0, stream>>>(
          h1, Wp2, b2, h2, 0, strips2 * 32, NC, NH);
    }
    if (remRows > 0) {
      gemm_bias_wmma<1><<<1, 32 * ng, 0, stream>>>(
          h1, Wp2, b2, h2, strips2 * 32, M, NC, NH);
    }
  }
  // ---- agg2 = h2 (self loop); agg2 += h2[src]
  {
    const long long n4 = (long long)M * NC / 4;
    copy_f4<<<(unsigned)((n4 + 255) / 256), 256, 0, stream>>>(
        (const float4*)h2, (float4*)agg2, n4);
    const long long nW = (long long)E * (NC / 4);
    scatter_edges<<<(unsigned)((nW + 255) / 256), 256, 0, stream>>>(
        srcI, dstI, h2, agg2, NC, 4, nW);
  }
  // ---- out = log_softmax(agg2 / cnt)
  mean_logsoftmax64<<<(M + 7) / 8, 256, 0, stream>>>(agg2, cnt, out, M);
}